// CTR_FM_83545703842340
// MI455X (gfx1250) — compile-verified
//
#include <hip/hip_runtime.h>

typedef __attribute__((ext_vector_type(2))) float v2f;
typedef __attribute__((ext_vector_type(8))) float v8f;

#define NUM_FIELDS 26
#define VOCAB      100000
#define EMBED_DIM  64
#define DENSE_DIM  13

// One wave (32 lanes) owns a tile of 16 samples.
// Lane L (<16) and lane L+16 cooperatively own sample L:
//   lane L    holds accumulated sum_f dims {4c, 4c+1}   (c = 0..15)
//   lane L+16 holds accumulated sum_f dims {4c+2, 4c+3}
// which is exactly the V_WMMA_F32_16X16X4_F32 A-matrix layout per K-chunk c.
// 16 chained WMMAs compute the 16x16 Gram matrix S*S^T over K=64; the
// diagonal is ||sum_f e||^2 per sample (exact f32, matches reference).
__global__ __launch_bounds__(256)
void ctr_fm_wmma_kernel(const int*   __restrict__ xs,
                        const float* __restrict__ xd,
                        const float* __restrict__ emb,
                        const float* __restrict__ lin,
                        const float* __restrict__ wd,
                        const float* __restrict__ bias,
                        float*       __restrict__ out,
                        int batch)
{
    const int lane = threadIdx.x & 31;
    const int wave = (blockIdx.x * blockDim.x + threadIdx.x) >> 5;
    const int s    = lane & 15;     // sample slot within tile
    const int hi   = lane >> 4;     // 0 -> dims {0,1} mod 4, 1 -> dims {2,3} mod 4
    const int sampleBase = wave * 16;
    const int sample     = sampleBase + s;
    if (sampleBase >= batch) return;   // whole-wave early-out only (EXEC stays all-1s)

    // sum_f accumulator, laid out as WMMA A-operand chunks (32 VGPRs)
    v2f acc[16];
#pragma unroll
    for (int c = 0; c < 16; ++c) { acc[c].x = 0.f; acc[c].y = 0.f; }
    float sq     = 0.f;   // per-lane partial of sum_{f,d} e^2 (this lane's 32 dims)
    float linear = 0.f;   // full per-sample lin-gather sum (identical on both halves)

    // Prefetch this tile's embedding rows: each lane pulls its 128B half-row
    // into cache ahead of the gather loop (lowers to global_prefetch_b8).
#pragma unroll
    for (int f = 0; f < NUM_FIELDS; ++f) {
        const int idx = xs[sample * NUM_FIELDS + f];
        const float* row = emb + ((size_t)f * VOCAB + (size_t)idx) * EMBED_DIM;
        __builtin_prefetch(row + hi * 32, 0, 0);
    }

    // Gather + accumulate. Per field each lane issues 16 x b64 loads covering
    // its 128B half of the 256B row, landing directly in WMMA A layout.
    for (int f = 0; f < NUM_FIELDS; ++f) {
        const int idx = xs[sample * NUM_FIELDS + f];
        const float* row = emb + ((size_t)f * VOCAB + (size_t)idx) * EMBED_DIM;
#pragma unroll
        for (int c = 0; c < 16; ++c) {
            v2f e = *(const v2f*)(row + c * 4 + hi * 2);
            acc[c].x += e.x;
            acc[c].y += e.y;
            sq = fmaf(e.x, e.x, sq);
            sq = fmaf(e.y, e.y, sq);
        }
        linear += lin[(size_t)f * VOCAB + (size_t)idx];
    }

    // Gram = S * S^T via 16 x V_WMMA_F32_16X16X4_F32 (K = 64).
    // For the symmetric product the A and B register images coincide.
    v8f gram = {0.f, 0.f, 0.f, 0.f, 0.f, 0.f, 0.f, 0.f};
#pragma unroll
    for (int c = 0; c < 16; ++c) {
        gram = __builtin_amdgcn_wmma_f32_16x16x4_f32(
            /*neg_a=*/false, acc[c],
            /*neg_b=*/false, acc[c],
            /*c_mod=*/(short)0, gram,
            /*reuse_a=*/false, /*reuse_b=*/false);
    }

    // Diagonal extraction per C/D layout:
    //   sample m<8  -> lane m,     VGPR m
    //   sample m>=8 -> lane m+16,  VGPR m-8
    float diag = 0.f;
    const int r = (lane < 8) ? lane : (lane - 24);
#pragma unroll
    for (int k = 0; k < 8; ++k) diag = (r == k) ? gram[k] : diag;

    const float diagX      = __shfl_xor(diag, 16, 32);  // bring hi-lane diagonal down
    const float sqX        = __shfl_xor(sq,   16, 32);
    const float square_sum = sq + sqX;                  // full sum_{f,d} e^2
    const float sum_square = (s < 8) ? diag : diagX;    // ||sum_f e||^2

    if (hi == 0 && sample < batch) {
        float dense = 0.f;
#pragma unroll
        for (int j = 0; j < DENSE_DIM; ++j)
            dense = fmaf(xd[sample * DENSE_DIM + j], wd[j], dense);
        const float fm = 0.5f * (sum_square - square_sum);
        out[sample] = linear + dense + bias[0] + fm;
    }
}

extern "C" void kernel_launch(void* const* d_in, const int* in_sizes, int n_in,
                              void* d_out, int out_size, void* d_ws, size_t ws_size,
                              hipStream_t stream) {
    const int*   xs   = (const int*)  d_in[0];   // x_sparse  [B, 26]
    const float* xd   = (const float*)d_in[1];   // x_dense   [B, 13]
    const float* emb  = (const float*)d_in[2];   // embed     [26, 100000, 64]
    const float* lin  = (const float*)d_in[3];   // lin_table [26, 100000]
    const float* wd   = (const float*)d_in[4];   // w_dense   [13]
    const float* bias = (const float*)d_in[5];   // bias      [1]
    float* out = (float*)d_out;

    const int batch = in_sizes[0] / NUM_FIELDS;        // 16384
    const int samplesPerBlock = 16 * (256 / 32);       // 16 per wave * 8 waves = 128
    const int grid = (batch + samplesPerBlock - 1) / samplesPerBlock;

    ctr_fm_wmma_kernel<<<grid, 256, 0, stream>>>(xs, xd, emb, lin, wd, bias, out, batch);
}